// GATv2WithNeighbors_68049461838097
// MI455X (gfx1250) — compile-verified
//
#include <hip/hip_runtime.h>
#include <hip/hip_bf16.h>

#define N_NODES 100000
#define E_EDGES 1600000
#define IN_DIM  20
#define HID     32
#define HEADS   4
#define OUT_CH  64

typedef __attribute__((ext_vector_type(2))) float v2f;
typedef __attribute__((ext_vector_type(8))) float v8f;

// ---------- helpers ----------
__device__ __forceinline__ float waveReduceSum(float v) {
#pragma unroll
  for (int off = 16; off > 0; off >>= 1) v += __shfl_xor(v, off, 32);
  return v;
}

// float atomic max via ordered-int trick (valid for any sign mix)
__device__ __forceinline__ void atomicMaxF(float* addr, float v) {
  if (v >= 0.0f) atomicMax((int*)addr, __float_as_int(v));
  else           atomicMin((unsigned int*)addr, (unsigned int)__float_as_int(v));
}

// ---------- small utility kernels ----------
__global__ void fill_kernel(float* __restrict__ p, float v, long n) {
  long i = (long)blockIdx.x * blockDim.x + threadIdx.x;
  if (i < n) p[i] = v;
}

__global__ void sum_kernel(const float* __restrict__ a, float* __restrict__ out, int n) {
  float s = 0.0f;
  for (int i = blockIdx.x * blockDim.x + threadIdx.x; i < n; i += gridDim.x * blockDim.x)
    s += a[i];
  s = waveReduceSum(s);
  if ((threadIdx.x & 31) == 0) atomicAdd(out, s);
}

__global__ void finalize_elu_kernel(float* __restrict__ a, const float* __restrict__ b,
                                    long n, int hc) {
  long i = (long)blockIdx.x * blockDim.x + threadIdx.x;
  if (i >= n) return;
  float v = a[i] + b[i % hc];
  a[i] = (v > 0.0f) ? v : (expf(v) - 1.0f);
}

__global__ void finalize_bias_kernel(float* __restrict__ a, const float* __restrict__ b,
                                     long n, int hc) {
  long i = (long)blockIdx.x * blockDim.x + threadIdx.x;
  if (i >= n) return;
  a[i] = a[i] + b[i % hc];
}

// ---------- WMMA f32 GEMM: C[M,NC] = A[M,K] * W[NC,K]^T + bias ----------
// One wave per 16-row strip; NT = NC/16 column tiles; K multiple of 4.
// V_WMMA_F32_16X16X4_F32: A 16x4 (lanes 0-15: K0/K1, lanes 16-31: K2/K3),
// B mirrored with N in place of M, C/D: VGPR j / lane l -> row j(+8), col l&15.
template<int K, int NC>
__global__ void gemm_xw_bias(const float* __restrict__ A, const float* __restrict__ W,
                             const float* __restrict__ bias, float* __restrict__ C, int M) {
  const int lane = threadIdx.x & 31;
  const int wave = blockIdx.x * (blockDim.x >> 5) + (threadIdx.x >> 5);
  const int row0 = wave << 4;
  if (row0 >= M) return;                 // wave-uniform: EXEC stays all-1s for WMMA
  const int m  = lane & 15;
  const int kh = (lane >> 4) << 1;       // 0 or 2
  constexpr int NT = NC / 16;
  const v8f vzero = {0.f, 0.f, 0.f, 0.f, 0.f, 0.f, 0.f, 0.f};
  v8f acc[NT];
#pragma unroll
  for (int t = 0; t < NT; ++t) acc[t] = vzero;
  const float* arow = A + (size_t)(row0 + m) * K;
  for (int k = 0; k < K; k += 4) {
    v2f a;
    a.x = arow[k + kh];
    a.y = arow[k + kh + 1];
#pragma unroll
    for (int t = 0; t < NT; ++t) {
      const int n = t * 16 + m;          // B[k,n] = W[n,k]
      v2f b;
      b.x = W[n * K + k + kh];
      b.y = W[n * K + k + kh + 1];
      acc[t] = __builtin_amdgcn_wmma_f32_16x16x4_f32(
          false, a, false, b, (short)0, acc[t], false, false);
    }
  }
  const int colL = lane & 15;
  const int rb   = row0 + ((lane >> 4) << 3);
#pragma unroll
  for (int t = 0; t < NT; ++t) {
    const int col = t * 16 + colL;
    const float bv = bias[col];
#pragma unroll
    for (int j = 0; j < 8; ++j)
      C[(size_t)(rb + j) * NC + col] = acc[t][j] + bv;
  }
}

// ---------- edge kernels (wave-per-edge) ----------
// logits[e,h] = sum_c att[h,c]*leakyrelu(xl[src]+xr[dst]+ea*We), + atomic segment max
template<int H, int C>
__global__ void edge_logits_kernel(const float* __restrict__ xl, const float* __restrict__ xr,
                                   const float* __restrict__ We, const float* __restrict__ att,
                                   const int* __restrict__ ei, const float* __restrict__ ea,
                                   const float* __restrict__ easum,
                                   float* __restrict__ logits, float* __restrict__ mx,
                                   int E, int N) {
  constexpr int HC = H * C;
  const int lane = threadIdx.x & 31;
  const int e = blockIdx.x * (blockDim.x >> 5) + (threadIdx.x >> 5);
  const int E2 = E + N;
  if (e >= E2) return;
  int src, dst; float eav;
  if (e < E) { src = ei[e]; dst = ei[E + e]; eav = ea[e]; }
  else       { src = dst = e - E;            eav = easum[0] * (1.0f / (float)E); }
  float acc[H];
#pragma unroll
  for (int h = 0; h < H; ++h) acc[h] = 0.0f;
#pragma unroll
  for (int i = 0; i < HC / 32; ++i) {
    const int hc = lane + 32 * i;
    float v = xl[(size_t)src * HC + hc] + xr[(size_t)dst * HC + hc] + eav * We[hc];
    v = (v > 0.0f) ? v : 0.2f * v;       // leaky relu
    acc[(32 * i) / C] += att[hc] * v;    // (32*i)/C is compile-time
  }
#pragma unroll
  for (int h = 0; h < H; ++h) {
    float s = waveReduceSum(acc[h]);
    if (lane == 0) {
      logits[(size_t)e * H + h] = s;
      atomicMaxF(&mx[(size_t)dst * H + h], s);
    }
  }
}

// ex = exp(logit - mx[dst]); store in place; denom[dst] += ex
template<int H>
__global__ void edge_exp_kernel(float* __restrict__ logits, const float* __restrict__ mx,
                                float* __restrict__ den, const int* __restrict__ ei,
                                int E, int N) {
  long t = (long)blockIdx.x * blockDim.x + threadIdx.x;
  const long total = (long)(E + N) * H;
  if (t >= total) return;
  const int e = (int)(t / H);
  const int h = (int)(t % H);
  const int dst = (e < E) ? ei[E + e] : (e - E);
  const float ex = expf(logits[t] - mx[(size_t)dst * H + h]);
  logits[t] = ex;
  atomicAdd(&den[(size_t)dst * H + h], ex);
}

// agg[dst] += xl[src] * (ex / denom[dst])
template<int H, int C>
__global__ void edge_agg_kernel(const float* __restrict__ xl, const float* __restrict__ ex,
                                const float* __restrict__ den, const int* __restrict__ ei,
                                float* __restrict__ agg, int E, int N) {
  constexpr int HC = H * C;
  const int lane = threadIdx.x & 31;
  const int e = blockIdx.x * (blockDim.x >> 5) + (threadIdx.x >> 5);
  const int E2 = E + N;
  if (e >= E2) return;
  int src, dst;
  if (e < E) { src = ei[e]; dst = ei[E + e]; }
  else       { src = dst = e - E; }
  float alpha[H];
#pragma unroll
  for (int h = 0; h < H; ++h)
    alpha[h] = ex[(size_t)e * H + h] / den[(size_t)dst * H + h];
#pragma unroll
  for (int i = 0; i < HC / 32; ++i) {
    const int hc = lane + 32 * i;
    atomicAdd(&agg[(size_t)dst * HC + hc],
              xl[(size_t)src * HC + hc] * alpha[(32 * i) / C]);
  }
}

// ---------- launch ----------
static inline int cdivl(long a, long b) { return (int)((a + b - 1) / b); }

extern "C" void kernel_launch(void* const* d_in, const int* in_sizes, int n_in,
                              void* d_out, int out_size, void* d_ws, size_t ws_size,
                              hipStream_t stream) {
  const float* x    = (const float*)d_in[0];
  const int*   ei   = (const int*)  d_in[1];
  const float* ea   = (const float*)d_in[2];
  const float* Wl1  = (const float*)d_in[3];
  const float* bl1  = (const float*)d_in[4];
  const float* Wr1  = (const float*)d_in[5];
  const float* br1  = (const float*)d_in[6];
  const float* We1  = (const float*)d_in[7];
  const float* att1 = (const float*)d_in[8];
  const float* b1   = (const float*)d_in[9];
  const float* Wl2  = (const float*)d_in[10];
  const float* bl2  = (const float*)d_in[11];
  const float* Wr2  = (const float*)d_in[12];
  const float* br2  = (const float*)d_in[13];
  const float* We2  = (const float*)d_in[14];
  const float* att2 = (const float*)d_in[15];
  const float* b2   = (const float*)d_in[16];
  float* out = (float*)d_out;

  const int N = N_NODES, E = E_EDGES, E2 = E + N;
  const int HC1 = HEADS * HID;   // 128

  float* ws = (float*)d_ws;
  size_t o = 0;
  float* xl    = ws + o; o += (size_t)N * HC1;   // layer2 reuses (N*64 fits)
  float* xr    = ws + o; o += (size_t)N * HC1;
  float* hb    = ws + o; o += (size_t)N * HC1;   // agg1 -> h (in-place ELU)
  float* lg    = ws + o; o += (size_t)E2 * HEADS;
  float* mx    = ws + o; o += (size_t)N * HEADS;
  float* dn    = ws + o; o += (size_t)N * HEADS;
  float* easum = ws + o; o += 64;

  const int TPB = 256;                         // 8 wave32 waves per block
  const int gemmBlocks = cdivl(N / 16, 8);     // wave per 16-row strip
  const int edgeBlocks = cdivl(E2, 8);         // wave per edge

  // edge_attr mean (self-loop fill value)
  fill_kernel<<<1, 64, 0, stream>>>(easum, 0.0f, 64);
  sum_kernel<<<256, TPB, 0, stream>>>(ea, easum, E);

  // ---- layer 1 ----
  gemm_xw_bias<IN_DIM, 128><<<gemmBlocks, TPB, 0, stream>>>(x, Wl1, bl1, xl, N);
  gemm_xw_bias<IN_DIM, 128><<<gemmBlocks, TPB, 0, stream>>>(x, Wr1, br1, xr, N);
  fill_kernel<<<cdivl((long)N * HEADS, TPB), TPB, 0, stream>>>(mx, -3.0e38f, (long)N * HEADS);
  fill_kernel<<<cdivl((long)N * HEADS, TPB), TPB, 0, stream>>>(dn, 0.0f, (long)N * HEADS);
  fill_kernel<<<cdivl((long)N * HC1, TPB), TPB, 0, stream>>>(hb, 0.0f, (long)N * HC1);
  edge_logits_kernel<HEADS, HID><<<edgeBlocks, TPB, 0, stream>>>(
      xl, xr, We1, att1, ei, ea, easum, lg, mx, E, N);
  edge_exp_kernel<HEADS><<<cdivl((long)E2 * HEADS, TPB), TPB, 0, stream>>>(lg, mx, dn, ei, E, N);
  edge_agg_kernel<HEADS, HID><<<edgeBlocks, TPB, 0, stream>>>(xl, lg, dn, ei, hb, E, N);
  finalize_elu_kernel<<<cdivl((long)N * HC1, TPB), TPB, 0, stream>>>(hb, b1, (long)N * HC1, HC1);

  // ---- layer 2 (heads=1, ch=64, no concat; aggregate straight into d_out) ----
  gemm_xw_bias<128, OUT_CH><<<gemmBlocks, TPB, 0, stream>>>(hb, Wl2, bl2, xl, N);
  gemm_xw_bias<128, OUT_CH><<<gemmBlocks, TPB, 0, stream>>>(hb, Wr2, br2, xr, N);
  fill_kernel<<<cdivl((long)N, TPB), TPB, 0, stream>>>(mx, -3.0e38f, (long)N);
  fill_kernel<<<cdivl((long)N, TPB), TPB, 0, stream>>>(dn, 0.0f, (long)N);
  fill_kernel<<<cdivl((long)N * OUT_CH, TPB), TPB, 0, stream>>>(out, 0.0f, (long)N * OUT_CH);
  edge_logits_kernel<1, OUT_CH><<<edgeBlocks, TPB, 0, stream>>>(
      xl, xr, We2, att2, ei, ea, easum, lg, mx, E, N);
  edge_exp_kernel<1><<<cdivl((long)E2, TPB), TPB, 0, stream>>>(lg, mx, dn, ei, E, N);
  edge_agg_kernel<1, OUT_CH><<<edgeBlocks, TPB, 0, stream>>>(xl, lg, dn, ei, out, E, N);
  finalize_bias_kernel<<<cdivl((long)N * OUT_CH, TPB), TPB, 0, stream>>>(
      out, b2, (long)N * OUT_CH, OUT_CH);
}